// FastLearnableEMA_27281632264619
// MI455X (gfx1250) — compile-verified
//
#include <hip/hip_runtime.h>
#include <stdint.h>

// Problem constants (from the reference): x,y = [B, T, C] fp32
#define T_TOTAL 2048
#define C_TOTAL 512
#define CH      32          // channels per workgroup (== wave width)
#define WAVES   8
#define SEG     (T_TOTAL / WAVES)   // 256 timesteps per wave
#define TPB     (WAVES * 32)        // 256 threads

// LDS: full x tile [2048 x 32] fp32 (256 KB) + per-wave segment sums (1 KB)
#define XS_FLOATS   (T_TOTAL * CH)
#define SMEM_FLOATS (XS_FLOATS + WAVES * CH)
#define SMEM_BYTES  (SMEM_FLOATS * sizeof(float))

typedef unsigned int u32x4 __attribute__((ext_vector_type(4)));
typedef int          i32x8 __attribute__((ext_vector_type(8)));
typedef int          i32x4 __attribute__((ext_vector_type(4)));

// Issue one TENSOR_LOAD_TO_LDS moving a [tile_rows x CH] fp32 tile whose rows
// are contiguous 128-byte runs strided by C_TOTAL*4 bytes in global memory,
// packed densely into LDS starting at lds_byte_off.
// D# layout per CDNA5 ISA ch.8 (group0 128b, group1 256b; remaining groups -> 0).
__device__ __forceinline__ void tdm_load_tile(uint32_t lds_byte_off,
                                              const void* gsrc,
                                              uint32_t tile_rows)
{
    uint64_t ga = (uint64_t)(uintptr_t)gsrc;

    u32x4 g0;
    g0[0] = 1u;                                   // count=1 (valid), user mode, no gather
    g0[1] = lds_byte_off;                         // lds_addr [63:32]
    g0[2] = (uint32_t)(ga & 0xFFFFFFFFull);       // global_addr[31:0]
    g0[3] = (uint32_t)((ga >> 32) & 0x1FFFFFFull) // global_addr[56:32]
          | (2u << 30);                           // type = 2 ("image") [127:126]

    const uint32_t tensor_dim0 = 1u << 20;        // generous: never clip OOB
    const uint32_t tensor_dim1 = 1u << 20;
    const uint32_t tile_dim0   = CH;              // 32 contiguous fp32 per row
    const uint32_t tile_dim1   = tile_rows;       // rows
    const uint64_t stride0     = C_TOTAL;         // row stride in elements

    i32x8 g1;
    g1[0] = (int)(2u << 16);                      // workgroup_mask=0, data_size=2 (4B)
    g1[1] = (int)((tensor_dim0 & 0xFFFFu) << 16); // tensor_dim0[15:0] -> [63:48]
    g1[2] = (int)(((tensor_dim0 >> 16) & 0xFFFFu)        // tensor_dim0[31:16]
          |       ((tensor_dim1 & 0xFFFFu) << 16));      // tensor_dim1[15:0]
    g1[3] = (int)(((tensor_dim1 >> 16) & 0xFFFFu)        // tensor_dim1[31:16]
          |       ((tile_dim0 & 0xFFFFu) << 16));        // tile_dim0 -> [127:112]
    g1[4] = (int)(tile_dim1 & 0xFFFFu);                  // tile_dim1; tile_dim2=0
    g1[5] = (int)(uint32_t)(stride0 & 0xFFFFFFFFull);    // tensor_dim0_stride[31:0]
    g1[6] = (int)(uint32_t)((stride0 >> 32) & 0xFFFFull);// stride[47:32]; dim1_stride lo=0
    g1[7] = 0;

    i32x4 z4 = {0, 0, 0, 0};
    i32x8 z8 = {0, 0, 0, 0, 0, 0, 0, 0};
    // clang-23 / therock-10.0 arity: (g0, g1, g2, g3, extra_group, cpol)
    __builtin_amdgcn_tensor_load_to_lds(g0, g1, z4, z4, z8, 0);
}

__global__ __launch_bounds__(TPB)
void FastLearnableEMA_kernel(const float* __restrict__ x,
                             const float* __restrict__ logit_alpha,
                             float* __restrict__ y)
{
    extern __shared__ float smem[];
    float* xs     = smem;                 // [T_TOTAL][CH]
    float* segsum = smem + XS_FLOATS;     // [WAVES][CH]

    const int b   = blockIdx.x >> 4;              // C_TOTAL/CH == 16 c-groups
    const int c0  = (blockIdx.x & 15) * CH;
    const int w   = threadIdx.x >> 5;             // wave id (wave32)
    const int c   = threadIdx.x & 31;             // channel lane
    const int t0  = w * SEG;

    // ---- per-wave async TDM: my [SEG x CH] x-tile -> LDS; wait only my own ----
    {
        const float* gsrc = x + ((size_t)b * T_TOTAL + t0) * C_TOTAL + c0;
        uint32_t lds_off  = (uint32_t)(uintptr_t)(&xs[t0 * CH]);
        tdm_load_tile(lds_off, gsrc, SEG);
        __builtin_amdgcn_s_wait_tensorcnt(0);
    }

    // ---- per-channel coefficient ----
    const float la  = logit_alpha[c0 + c];
    float a = 1.0f / (1.0f + expf(-la));
    a = fminf(fmaxf(a, 1e-4f), 1.0f - 1e-4f);
    const float oma  = 1.0f - a;
    const float inva = 1.0f / a;

    // scale[] factored out of all scans:
    //   S_t = oma * A_t + a * x0,  A_t = sum_{u<=t} x_u a^u   (coeff of x0: oma+a=1 ok)
    // so both phases run the bare geometric cumsum A (2 VALU/element).

    // ---- phase 1: unweighted segment sum  L_w = sum_j x_j a^j (local scale) ----
    {
        float ap = 1.0f, local = 0.0f;
        #pragma unroll 8
        for (int j = 0; j < SEG; ++j) {
            local = fmaf(xs[(t0 + j) * CH + c], ap, local);
            ap *= a;
        }
        segsum[w * CH + c] = local;
    }
    __syncthreads();

    // ---- cross-wave exclusive prefix (global power scale) ----
    const float a256 = powf(a, (float)SEG);
    float carry = 0.0f, apg = 1.0f;               // apg ends as a^{t0}
    for (int v = 0; v < w; ++v) {
        carry = fmaf(apg, segsum[v * CH + c], carry);
        apg *= a256;
    }

    // x[b,0,c] for this block's channels (wave 0's TDM segment, visible post-barrier)
    const float K = a * xs[c];

    // ---- phase 2: final scan; y = (oma*A_t + K) * min(a^-t, 1/EPS) ----
    float acc = carry;
    float ap  = apg;
    float ip  = 1.0f / apg;                       // inf on underflow -> clamped below
    float* ydst = y + ((size_t)b * T_TOTAL + t0) * C_TOTAL + c0 + c;
    #pragma unroll 8
    for (int j = 0; j < SEG; ++j) {
        acc = fmaf(xs[(t0 + j) * CH + c], ap, acc);
        float s   = fmaf(oma, acc, K);
        float out = s * fminf(ip, 1e8f);
        __builtin_nontemporal_store(out, ydst + (size_t)j * C_TOTAL);
        ap *= a;
        ip *= inva;
    }
}

extern "C" void kernel_launch(void* const* d_in, const int* in_sizes, int n_in,
                              void* d_out, int out_size, void* d_ws, size_t ws_size,
                              hipStream_t stream) {
    const float* x  = (const float*)d_in[0];
    const float* la = (const float*)d_in[1];
    float* y        = (float*)d_out;

    const int B = in_sizes[0] / (T_TOTAL * C_TOTAL);   // 32

    // Opt in to >48KB dynamic LDS (263 KB tile; WGP has 320 KB).
    (void)hipFuncSetAttribute((const void*)FastLearnableEMA_kernel,
                              hipFuncAttributeMaxDynamicSharedMemorySize,
                              (int)SMEM_BYTES);

    dim3 grid(B * (C_TOTAL / CH));   // 512 workgroups x 8 waves = 4096 waves
    dim3 block(TPB);
    FastLearnableEMA_kernel<<<grid, block, SMEM_BYTES, stream>>>(x, la, y);
}